// GAT_17600775979469
// MI455X (gfx1250) — compile-verified
//
#include <hip/hip_runtime.h>
#include <hip/hip_bf16.h>
#include <math.h>

typedef __attribute__((ext_vector_type(2))) float v2f;
typedef __attribute__((ext_vector_type(8))) float v8f;

#define CDIV(a, b) (((a) + (b) - 1) / (b))

// ---------------- utility ----------------

__global__ void fillK(float* __restrict__ p, float v, int n) {
    int i = blockIdx.x * blockDim.x + threadIdx.x;
    if (i < n) p[i] = v;
}

__device__ __forceinline__ void atomicMaxFloat(float* addr, float val) {
    // signed-max for positives, unsigned-min for negatives; init must be -inf
    if (val >= 0.0f) {
        atomicMax((int*)addr, __float_as_int(val));
    } else {
        atomicMin((unsigned int*)addr, (unsigned int)__float_as_int(val));
    }
}

// ---------------- preprocessing: self-loop edge_attr mean ----------------

__global__ void countK(const int* __restrict__ dst, const float* __restrict__ edge_attr,
                       float* __restrict__ cnt, float* __restrict__ ea_sum, int E) {
    int e = blockIdx.x * blockDim.x + threadIdx.x;
    if (e >= E) return;
    int d = dst[e];
    atomicAdd(cnt + d, 1.0f);
    const float* ea = edge_attr + (size_t)e * 12;
    float* out = ea_sum + (size_t)d * 12;
    #pragma unroll
    for (int j = 0; j < 12; ++j) atomicAdd(out + j, ea[j]);
}

__global__ void eaDivK(float* __restrict__ ea_loop, const float* __restrict__ cnt, int N) {
    int t = blockIdx.x * blockDim.x + threadIdx.x;
    if (t >= N * 12) return;
    int n = t / 12;
    ea_loop[t] = ea_loop[t] / fmaxf(cnt[n], 1.0f);
}

// ---------------- layer 0 linear: [N,13] @ [13,64] ----------------

__global__ void lin13K(const float* __restrict__ x, const float* __restrict__ W0,
                       float* __restrict__ h, int N) {
    int t = blockIdx.x * blockDim.x + threadIdx.x;
    if (t >= N * 64) return;
    int n = t >> 6, c = t & 63;
    const float* xr = x + (size_t)n * 13;
    float s = 0.0f;
    #pragma unroll
    for (int k = 0; k < 13; ++k) s = fmaf(xr[k], W0[k * 64 + c], s);
    h[t] = s;
}

// ---------------- layers 1..4 linear: [N,64] @ [64,64] via V_WMMA_F32_16X16X4_F32 ----
// One wave per 16x16 output tile. ISA 7.12.2 32-bit A 16x4 layout:
//   lane L: row = L&15, K-offset = (L>>4)*2 ; A frag = {A[row][k+off], A[row][k+off+1]}
// B 4x16 symmetric: col = L&15, K row pair = (L>>4)*2.
// C/D 16x16: VGPR r holds (M = r + 8*(L>=16), Nc = L&15).

__global__ void wmmaLin64K(const float* __restrict__ A, const float* __restrict__ W,
                           float* __restrict__ Hout, int N) {
    int lane = threadIdx.x;            // 0..31
    int r = lane & 15;
    int koff = (lane >> 4) << 1;       // 0 or 2
    int rowBase = blockIdx.x * 16;
    int colBase = blockIdx.y * 16;
    const float* arow = A + (size_t)(rowBase + r) * 64;
    const float* bcol = W + colBase + r;
    v8f acc = {};
    #pragma unroll
    for (int k = 0; k < 64; k += 4) {
        v2f a, b;
        a.x = arow[k + koff];
        a.y = arow[k + koff + 1];
        b.x = bcol[(size_t)(k + koff) * 64];
        b.y = bcol[(size_t)(k + koff + 1) * 64];
        acc = __builtin_amdgcn_wmma_f32_16x16x4_f32(
            /*neg_a=*/false, a, /*neg_b=*/false, b,
            /*c_mod=*/(short)0, acc, /*reuse_a=*/false, /*reuse_b=*/false);
    }
    int mBase = rowBase + ((lane >> 4) << 3);
    #pragma unroll
    for (int i = 0; i < 8; ++i) {
        Hout[(size_t)(mBase + i) * 64 + colBase + r] = acc[i];
    }
}

// ---------------- per-edge attention weight vector: we_vec = W_edge[l] @ att_edge[l] ---

__global__ void weVecK(const float* __restrict__ We, const float* __restrict__ ae,
                       float* __restrict__ wv) {
    int d = threadIdx.x;
    if (d >= 12) return;
    const float* row = We + (size_t)d * 64;
    float s = 0.0f;
    #pragma unroll
    for (int h = 0; h < 64; ++h) s = fmaf(row[h], ae[h], s);
    wv[d] = s;
}

// ---------------- node attention scalars ----------------

__global__ void attnScalK(const float* __restrict__ h, const float* __restrict__ a_s,
                          const float* __restrict__ a_d, float* __restrict__ s_src,
                          float* __restrict__ s_dst, int N) {
    int n = blockIdx.x * blockDim.x + threadIdx.x;
    if (n >= N) return;
    const float* hr = h + (size_t)n * 64;
    float ss = 0.0f, sd = 0.0f;
    #pragma unroll
    for (int c = 0; c < 64; ++c) {
        float v = hr[c];
        ss = fmaf(v, a_s[c], ss);
        sd = fmaf(v, a_d[c], sd);
    }
    s_src[n] = ss;
    s_dst[n] = sd;
}

// ---------------- alpha = leaky_relu(s_src[s] + s_dst[d] + ea.we) ; segment max ------

__global__ void alphaK(const int* __restrict__ src, const int* __restrict__ dst,
                       const float* __restrict__ edge_attr, const float* __restrict__ ea_loop,
                       const float* __restrict__ we_vec, const float* __restrict__ s_src,
                       const float* __restrict__ s_dst, float* __restrict__ alpha,
                       float* __restrict__ m, int E, int N) {
    int e = blockIdx.x * blockDim.x + threadIdx.x;
    if (e >= E + N) return;
    int s, d;
    const float* ea;
    if (e < E) {
        s = src[e]; d = dst[e];
        ea = edge_attr + (size_t)e * 12;
    } else {
        s = d = e - E;
        ea = ea_loop + (size_t)(e - E) * 12;
    }
    float t = s_src[s] + s_dst[d];
    #pragma unroll
    for (int j = 0; j < 12; ++j) t = fmaf(ea[j], we_vec[j], t);
    t = (t > 0.0f) ? t : 0.2f * t;
    alpha[e] = t;
    atomicMaxFloat(m + d, t);
}

// ---------------- p = exp(alpha - m[d]) ; segment sum (in-place over alpha) ----------

__global__ void expK(const int* __restrict__ dst, float* __restrict__ alpha,
                     const float* __restrict__ m, float* __restrict__ denom, int E, int N) {
    int e = blockIdx.x * blockDim.x + threadIdx.x;
    if (e >= E + N) return;
    int d = (e < E) ? dst[e] : (e - E);
    float p = __expf(alpha[e] - m[d]);
    alpha[e] = p;
    atomicAdd(denom + d, p);
}

// ---------------- out init to bias ----------------

__global__ void initOutK(float* __restrict__ out, const float* __restrict__ bias, int N) {
    int t = blockIdx.x * blockDim.x + threadIdx.x;
    if (t >= N * 64) return;
    out[t] = bias[t & 63];
}

// ---------------- out[d] += coef * h[s]  (16 threads per edge, 4 cols each) ----------

__global__ void aggK(const int* __restrict__ src, const int* __restrict__ dst,
                     const float* __restrict__ p, const float* __restrict__ denom,
                     const float* __restrict__ h, float* __restrict__ out, int E, int N) {
    int tid = blockIdx.x * blockDim.x + threadIdx.x;
    int e = tid >> 4;
    if (e >= E + N) return;
    int c = (tid & 15) << 2;
    int s, d;
    if (e < E) { s = src[e]; d = dst[e]; }
    else       { s = d = e - E; }
    float coef = p[e] / denom[d];
    float4 v = *(const float4*)(h + (size_t)s * 64 + c);
    float* o = out + (size_t)d * 64 + c;
    atomicAdd(o + 0, coef * v.x);
    atomicAdd(o + 1, coef * v.y);
    atomicAdd(o + 2, coef * v.z);
    atomicAdd(o + 3, coef * v.w);
}

// ---------------- relu ----------------

__global__ void reluK(float* __restrict__ h, int n) {
    int t = blockIdx.x * blockDim.x + threadIdx.x;
    if (t < n) h[t] = fmaxf(h[t], 0.0f);
}

// ---------------- global mean pool + final linear ----------------

__global__ void poolK(const float* __restrict__ h, const int* __restrict__ batch,
                      float* __restrict__ pool_sum, float* __restrict__ gcnt, int N) {
    int t = blockIdx.x * blockDim.x + threadIdx.x;
    if (t >= N * 64) return;
    int n = t >> 6, c = t & 63;
    int g = batch[n];
    atomicAdd(pool_sum + (size_t)g * 64 + c, h[t]);
    if (c == 0) atomicAdd(gcnt + g, 1.0f);
}

__global__ void finalK(const float* __restrict__ pool_sum, const float* __restrict__ gcnt,
                       const float* __restrict__ lin_w, const float* __restrict__ lin_b,
                       float* __restrict__ out, int G) {
    int g = blockIdx.x * blockDim.x + threadIdx.x;
    if (g >= G) return;
    float inv = 1.0f / fmaxf(gcnt[g], 1.0f);
    const float* pr = pool_sum + (size_t)g * 64;
    float s = 0.0f;
    #pragma unroll
    for (int c = 0; c < 64; ++c) s = fmaf(pr[c] * inv, lin_w[c], s);
    out[g] = s + lin_b[0];
}

// ---------------- launch ----------------

extern "C" void kernel_launch(void* const* d_in, const int* in_sizes, int n_in,
                              void* d_out, int out_size, void* d_ws, size_t ws_size,
                              hipStream_t stream) {
    const int IN = 13, H = 64, ED = 12, L = 5, G = 64;
    const int N = in_sizes[0] / IN;     // 100000
    const int E = in_sizes[1] / 2;      // 1600000
    const int EN = E + N;

    const float* x        = (const float*)d_in[0];
    const int*   ei       = (const int*)d_in[1];
    const int*   src      = ei;
    const int*   dst      = ei + E;
    const float* edge_attr= (const float*)d_in[2];
    const int*   batch    = (const int*)d_in[3];
    const float* W0       = (const float*)d_in[4];
    const float* Ws       = (const float*)d_in[5];
    const float* att_src  = (const float*)d_in[6];
    const float* att_dst  = (const float*)d_in[7];
    const float* W_edge   = (const float*)d_in[8];
    const float* att_edge = (const float*)d_in[9];
    const float* bias     = (const float*)d_in[10];
    const float* lin_w    = (const float*)d_in[11];
    const float* lin_b    = (const float*)d_in[12];
    float* out = (float*)d_out;

    // workspace layout (floats); all offsets stay 16B aligned
    float* w = (float*)d_ws;
    float* cnt     = w; w += N;
    float* ea_loop = w; w += (size_t)N * ED;
    float* h_lin   = w; w += (size_t)N * H;
    float* h_act   = w; w += (size_t)N * H;
    float* s_src   = w; w += N;
    float* s_dst   = w; w += N;
    float* alpha   = w; w += EN;        // reused as p
    float* m       = w; w += N;
    float* denom   = w; w += N;
    float* we_vec  = w; w += 16;
    float* pool_s  = w; w += G * H;
    float* gcnt    = w; w += G;

    const int B = 256;

    // self-loop edge_attr mean
    fillK<<<CDIV(N, B), B, 0, stream>>>(cnt, 0.0f, N);
    fillK<<<CDIV(N * ED, B), B, 0, stream>>>(ea_loop, 0.0f, N * ED);
    countK<<<CDIV(E, B), B, 0, stream>>>(dst, edge_attr, cnt, ea_loop, E);
    eaDivK<<<CDIV(N * ED, B), B, 0, stream>>>(ea_loop, cnt, N);

    for (int l = 0; l < L; ++l) {
        // node linear transform
        if (l == 0) {
            lin13K<<<CDIV(N * H, B), B, 0, stream>>>(x, W0, h_lin, N);
        } else {
            dim3 grid(N / 16, H / 16);
            wmmaLin64K<<<grid, 32, 0, stream>>>(h_act, Ws + (size_t)(l - 1) * H * H, h_lin, N);
        }
        // attention terms
        weVecK<<<1, 16, 0, stream>>>(W_edge + (size_t)l * ED * H, att_edge + (size_t)l * H, we_vec);
        attnScalK<<<CDIV(N, B), B, 0, stream>>>(h_lin, att_src + (size_t)l * H,
                                                att_dst + (size_t)l * H, s_src, s_dst, N);
        fillK<<<CDIV(N, B), B, 0, stream>>>(m, -INFINITY, N);
        fillK<<<CDIV(N, B), B, 0, stream>>>(denom, 0.0f, N);
        alphaK<<<CDIV(EN, B), B, 0, stream>>>(src, dst, edge_attr, ea_loop, we_vec,
                                              s_src, s_dst, alpha, m, E, N);
        expK<<<CDIV(EN, B), B, 0, stream>>>(dst, alpha, m, denom, E, N);
        // aggregate
        initOutK<<<CDIV(N * H, B), B, 0, stream>>>(h_act, bias + (size_t)l * H, N);
        aggK<<<CDIV(EN * 16, B), B, 0, stream>>>(src, dst, alpha, denom, h_lin, h_act, E, N);
        if (l < L - 1) reluK<<<CDIV(N * H, B), B, 0, stream>>>(h_act, N * H);
    }

    // global mean pool + final linear
    fillK<<<CDIV(G * H, B), B, 0, stream>>>(pool_s, 0.0f, G * H);
    fillK<<<1, G, 0, stream>>>(gcnt, 0.0f, G);
    poolK<<<CDIV(N * H, B), B, 0, stream>>>(h_act, batch, pool_s, gcnt, N);
    finalK<<<1, G, 0, stream>>>(pool_s, gcnt, lin_w, lin_b, out, G);
}